// Soft_NN_74929999446257
// MI455X (gfx1250) — compile-verified
//
#include <hip/hip_runtime.h>

typedef __attribute__((ext_vector_type(2))) float v2f;
typedef __attribute__((ext_vector_type(8))) float v8f;

#define BATCH 4
#define NQ 4096
#define NM 4096
#define CDIM 32
#define WAVES_PER_BLOCK 8

// Pass 1: yn2[b*m] = 0.5 * ||y[b,m,:]||^2  (folded into WMMA accumulator init)
__global__ __launch_bounds__(256) void Soft_NN_yn2_kernel(
    const float* __restrict__ yf, float* __restrict__ yn2, int total) {
  int i = blockIdx.x * blockDim.x + threadIdx.x;
  if (i >= total) return;
  const float* p = yf + (size_t)i * CDIM;
  float s = 0.0f;
#pragma unroll
  for (int k = 0; k < CDIM; ++k) s += p[k] * p[k];
  yn2[i] = 0.5f * s;
}

// Issue the 8 B-fragment loads (+ accumulator seed) for one 16-column m-tile.
__device__ __forceinline__ void load_btile(const float* __restrict__ yb,
                                           const float* __restrict__ yn2b,
                                           int m0, int sub, int kHalf,
                                           v2f* __restrict__ bf, float& cinit) {
  const int col = m0 + sub;
  const float* yrow = yb + (size_t)col * CDIM;
#pragma unroll
  for (int ks = 0; ks < 8; ++ks)
    bf[ks] = *(const v2f*)(yrow + ks * 4 + kHalf);
  cinit = -yn2b[col];
}

// 8 WMMAs on a resident B tile: two independent accumulation chains (depth 4),
// then combine, then running-argmax update for this lane's column.
__device__ __forceinline__ void compute_tile(const v2f* __restrict__ afrag,
                                             const v2f* __restrict__ bf,
                                             float cinit, int m0, int sub,
                                             float* __restrict__ bestVal,
                                             int* __restrict__ bestIdx) {
  v8f acc0 = {cinit, cinit, cinit, cinit, cinit, cinit, cinit, cinit};
  v8f acc1 = {0.f, 0.f, 0.f, 0.f, 0.f, 0.f, 0.f, 0.f};
#pragma unroll
  for (int ks = 0; ks < 8; ks += 2) {
    acc0 = __builtin_amdgcn_wmma_f32_16x16x4_f32(
        false, afrag[ks], false, bf[ks], (short)0, acc0, false, false);
    acc1 = __builtin_amdgcn_wmma_f32_16x16x4_f32(
        false, afrag[ks + 1], false, bf[ks + 1], (short)0, acc1, false, false);
  }
  const int col = m0 + sub;
#pragma unroll
  for (int v = 0; v < 8; ++v) {
    float s = acc0[v] + acc1[v];
    if (s > bestVal[v]) { bestVal[v] = s; bestIdx[v] = col; }
  }
}

__global__ __launch_bounds__(32 * WAVES_PER_BLOCK) void Soft_NN_74929999446257_kernel(
    const float* __restrict__ xf, const float* __restrict__ yf,
    const float* __restrict__ yc, const float* __restrict__ yn2,
    float* __restrict__ out) {
  const int lane = threadIdx.x & 31;
  const int wave = threadIdx.x >> 5;
  const int tile = blockIdx.x * WAVES_PER_BLOCK + wave;  // global 16-row tile
  const int tilesPerBatch = NQ / 16;
  const int b = tile / tilesPerBatch;
  const int row0 = (tile % tilesPerBatch) * 16;

  const float* xb   = xf  + (size_t)b * NQ * CDIM;
  const float* yb   = yf  + (size_t)b * NM * CDIM;
  const float* yn2b = yn2 + (size_t)b * NM;

  const int halfSel = lane >> 4;   // 0: lanes 0-15 (rows 0-7), 1: lanes 16-31 (rows 8-15)
  const int sub     = lane & 15;
  const int kHalf   = halfSel * 2; // K offset within each 4-wide K step

  // A fragments for the whole row tile: row = row0+sub, K = 4*ks + kHalf + {0,1}
  v2f afrag[8];
#pragma unroll
  for (int ks = 0; ks < 8; ++ks)
    afrag[ks] = *(const v2f*)(xb + (size_t)(row0 + sub) * CDIM + ks * 4 + kHalf);

  float bestVal[8];
  int   bestIdx[8];
#pragma unroll
  for (int v = 0; v < 8; ++v) { bestVal[v] = -3.4e38f; bestIdx[v] = 0; }

  // Register double-buffered pipeline over m-tiles, 2 tiles per iteration.
  v2f bufA[8], bufB[8];
  float cA, cB;
  load_btile(yb, yn2b, 0, sub, kHalf, bufA, cA);

  for (int m0 = 0; m0 < NM; m0 += 32) {
    // Issue loads for tile t+1 while WMMAs chew on tile t (and vice versa).
    load_btile(yb, yn2b, m0 + 16, sub, kHalf, bufB, cB);
    compute_tile(afrag, bufA, cA, m0, sub, bestVal, bestIdx);

    if (m0 + 32 < NM) {
      load_btile(yb, yn2b, m0 + 32, sub, kHalf, bufA, cA);
      __builtin_prefetch(yb + (size_t)(m0 + 48 + sub) * CDIM, 0, 3);
    }
    compute_tile(afrag, bufB, cB, m0 + 16, sub, bestVal, bestIdx);
  }

  // Argmax reduction across the 16 lanes of each half-wave (ties -> lower m).
#pragma unroll
  for (int v = 0; v < 8; ++v) {
#pragma unroll
    for (int mask = 1; mask < 16; mask <<= 1) {
      float ov = __shfl_xor(bestVal[v], mask, 32);
      int   oi = __shfl_xor(bestIdx[v], mask, 32);
      if (ov > bestVal[v] || (ov == bestVal[v] && oi < bestIdx[v])) {
        bestVal[v] = ov; bestIdx[v] = oi;
      }
    }
  }

  // Lane 0 holds rows row0..row0+7; lane 16 holds rows row0+8..row0+15.
  if (sub == 0) {
#pragma unroll
    for (int v = 0; v < 8; ++v) {
      const int row = row0 + halfSel * 8 + v;
      const float* src = yc  + ((size_t)b * NM + bestIdx[v]) * 3;
      float*       dst = out + ((size_t)b * NQ + row) * 3;
      dst[0] = src[0]; dst[1] = src[1]; dst[2] = src[2];
    }
  }
}

extern "C" void kernel_launch(void* const* d_in, const int* in_sizes, int n_in,
                              void* d_out, int out_size, void* d_ws, size_t ws_size,
                              hipStream_t stream) {
  const float* x_f = (const float*)d_in[0];  // [4,4096,32]
  const float* y_f = (const float*)d_in[1];  // [4,4096,32]
  const float* y_c = (const float*)d_in[2];  // [4,4096,3]
  // d_in[3] = temp_inv: irrelevant to the forward value (argmax of softmax == argmin dist)
  float* out = (float*)d_out;                // [4,4096,3]
  float* yn2 = (float*)d_ws;                 // 4*4096 floats = 64 KB scratch

  const int total = BATCH * NM;
  Soft_NN_yn2_kernel<<<(total + 255) / 256, 256, 0, stream>>>(y_f, yn2, total);

  const int tiles  = BATCH * (NQ / 16);      // 1024 waves
  const int blocks = tiles / WAVES_PER_BLOCK;
  Soft_NN_74929999446257_kernel<<<blocks, 32 * WAVES_PER_BLOCK, 0, stream>>>(
      x_f, y_f, y_c, yn2, out);
}